// CTCBeamSearchDecoder_63642825392580
// MI455X (gfx1250) — compile-verified
//
#include <hip/hip_runtime.h>
#include <hip/hip_bf16.h>
#include <math.h>

#define B_    256
#define T_    1024
#define C_    92
#define CP_   96
#define BEAM_ 10
#define BLANK_ 91
#define KC_   (BEAM_ * C_)     // 920
#define NEG_  (-1e30f)
#define P1_   1000003u
#define P2_   2654435761u
#define OFF2_ 40503u
#define NTHREADS 256

typedef __attribute__((ext_vector_type(2))) float v2f;
typedef __attribute__((ext_vector_type(8))) float v8f;

__device__ __forceinline__ bool cand_better(float av, unsigned ah2, unsigned ah1, int ak,
                                            float bv, unsigned bh2, unsigned bh1, int bk) {
    if (av > bv) return true;
    if (av < bv) return false;
    if (ah2 < bh2) return true;
    if (ah2 > bh2) return false;
    if (ah1 < bh1) return true;
    if (ah1 > bh1) return false;
    return ak < bk;
}

__global__ __launch_bounds__(NTHREADS)
void ctc_beam_kernel(const float* __restrict__ logits,
                     const int*   __restrict__ lengths,
                     float*       __restrict__ out)
{
    const int b    = blockIdx.x;
    const int tid  = threadIdx.x;
    const int lane = tid & 31;
    const int wid  = tid >> 5;

    __shared__ float           sLogits[2][CP_];     // async-load target (double buffer)
    __shared__ float           sLp[CP_];            // log-softmax of current row (padded w/ 0)
    __shared__ float           sScore[16];
    __shared__ unsigned        sH1[16], sH2[16];
    __shared__ int             sLen[16], sLast[16];
    __shared__ float           nScore[16];
    __shared__ unsigned        nH1[16], nH2[16];
    __shared__ int             nLen[16], nLast[16];
    __shared__ float           sCs[KC_];            // raw candidate scores
    __shared__ float           sFs[KC_];            // dedupe-masked scores
    __shared__ unsigned long long sHH[KC_];         // (h2<<32)|h1
    __shared__ unsigned short  sBp[T_ * BEAM_];     // backpointers: parent|chg<<4|char<<5
    __shared__ float           wV[8];
    __shared__ unsigned        wL[8], wH[8];
    __shared__ int             wK[8];
    __shared__ int             selK[BEAM_];
    __shared__ float           selV[BEAM_];
    __shared__ int             sBest, sBlen;

    int len = lengths[b];
    if (len < 0) len = 0;
    if (len > T_) len = T_;

    if (tid < 16) {
        sScore[tid] = (tid == 0) ? 0.0f : NEG_;
        sH1[tid] = 1u; sH2[tid] = 1u;
        sLen[tid] = 0; sLast[tid] = 0;
    }
    __syncthreads();

    const float* rowbase = logits + (size_t)b * T_ * C_;

    // ---- prefetch row 0 (92 floats = 23 x b128, row stride 368B is 16B aligned) ----
    if (len > 0 && tid < 23) {
        unsigned lds = (unsigned)(size_t)(&sLogits[0][tid * 4]);
        unsigned long long ga = (unsigned long long)(size_t)(rowbase + (size_t)0 * C_ + tid * 4);
        asm volatile("global_load_async_to_lds_b128 %0, %1, off" :: "v"(lds), "v"(ga) : "memory");
    }

    for (int t = 0; t < len; ++t) {
        const int cur = t & 1;

        // issue prefetch of next row into the other buffer
        if (t + 1 < len && tid < 23) {
            unsigned lds = (unsigned)(size_t)(&sLogits[cur ^ 1][tid * 4]);
            unsigned long long ga = (unsigned long long)(size_t)(rowbase + (size_t)(t + 1) * C_ + tid * 4);
            asm volatile("global_load_async_to_lds_b128 %0, %1, off" :: "v"(lds), "v"(ga) : "memory");
        }
        if (wid == 0) {   // only wave 0 has outstanding async ops; loads complete in order
            if (t + 1 < len) asm volatile("s_wait_asynccnt 1" ::: "memory");
            else             asm volatile("s_wait_asynccnt 0" ::: "memory");
        }
        __syncthreads();

        // ---- wave 0: log_softmax of current row into sLp ----
        if (wid == 0) {
            float x0 = sLogits[cur][lane];
            float x1 = sLogits[cur][lane + 32];
            float x2 = (lane < 28) ? sLogits[cur][lane + 64] : -INFINITY;
            float m = fmaxf(fmaxf(x0, x1), x2);
            #pragma unroll
            for (int off = 16; off; off >>= 1) m = fmaxf(m, __shfl_xor(m, off, 32));
            float s = expf(x0 - m) + expf(x1 - m) + ((lane < 28) ? expf(x2 - m) : 0.0f);
            #pragma unroll
            for (int off = 16; off; off >>= 1) s += __shfl_xor(s, off, 32);
            float ls = m + logf(s);
            sLp[lane]      = x0 - ls;
            sLp[lane + 32] = x1 - ls;
            if (lane < 28) sLp[lane + 64] = x2 - ls;
            if (lane < 4)  sLp[92 + lane] = 0.0f;   // pad for WMMA tile 5
        }
        __syncthreads();

        // ---- candidate scores cs[p][c] = scores[p] + lp[c] via V_WMMA_F32_16X16X4_F32 ----
        // A(16x4) = [scores | 1 | 0 | 0], B(4x16) = [[1],[lp tile],[0],[0]]  (bit-exact f32)
#if __has_builtin(__builtin_amdgcn_wmma_f32_16x16x4_f32)
        if (wid < 6) {
            const int half = lane >> 4;       // 0: K0/K1 , 1: K2/K3
            const int mn   = lane & 15;
            const int n0   = wid * 16;
            v2f a, bm;
            if (!half) { a.x = sScore[mn]; a.y = 1.0f; } else { a.x = 0.0f; a.y = 0.0f; }
            if (!half) { bm.x = 1.0f; bm.y = sLp[n0 + mn]; } else { bm.x = 0.0f; bm.y = 0.0f; }
            v8f cacc = {};
            v8f d = __builtin_amdgcn_wmma_f32_16x16x4_f32(false, a, false, bm,
                                                          (short)0, cacc, false, false);
            #pragma unroll
            for (int r = 0; r < 8; ++r) {
                int m = r + 8 * half;
                int n = n0 + mn;
                if (m < BEAM_ && n < C_) sCs[m * C_ + n] = d[r];
            }
        }
#else
        for (int k = tid; k < KC_; k += NTHREADS) {
            int p = k / C_, c = k - p * C_;
            sCs[k] = sScore[p] + sLp[c];
        }
#endif

        // ---- hashes (all threads) ----
        for (int k = tid; k < KC_; k += NTHREADS) {
            int p = k / C_, c = k - p * C_;
            bool app = (c != BLANK_) && (sLen[p] == 0 || sLast[p] != c);
            unsigned cu = (unsigned)(c + 1);
            unsigned h1 = app ? (sH1[p] * P1_ + cu)         : sH1[p];
            unsigned h2 = app ? (sH2[p] * P2_ + cu * OFF2_) : sH2[p];
            sHH[k] = ((unsigned long long)h2 << 32) | (unsigned long long)h1;
        }
        __syncthreads();

        // ---- dedupe: keep max-score (tie: min index) per hash ----
        for (int k = tid; k < KC_; k += NTHREADS) {
            const unsigned long long hk = sHH[k];
            const float ck = sCs[k];
            bool dead = false;
            for (int j = 0; j < KC_; ++j) {
                if (sHH[j] == hk && j != k) {
                    float cj = sCs[j];
                    if (cj > ck || (cj == ck && j < k)) { dead = true; break; }
                }
            }
            sFs[k] = dead ? NEG_ : ck;
        }
        __syncthreads();

        // ---- top-BEAM selection, reference tie-order: value desc, (h2,h1) asc, k asc ----
        for (int r = 0; r < BEAM_; ++r) {
            float bv = -INFINITY; unsigned bl = 0xffffffffu, bh = 0xffffffffu; int bk = KC_;
            for (int k = tid; k < KC_; k += NTHREADS) {
                float v = sFs[k];
                unsigned long long h = sHH[k];
                unsigned hl = (unsigned)h, hh = (unsigned)(h >> 32);
                if (cand_better(v, hh, hl, k, bv, bh, bl, bk)) { bv = v; bl = hl; bh = hh; bk = k; }
            }
            #pragma unroll
            for (int off = 16; off; off >>= 1) {
                float    ov = __shfl_down(bv, off, 32);
                unsigned ol = __shfl_down(bl, off, 32);
                unsigned oh = __shfl_down(bh, off, 32);
                int      ok = __shfl_down(bk, off, 32);
                if (cand_better(ov, oh, ol, ok, bv, bh, bl, bk)) { bv = ov; bl = ol; bh = oh; bk = ok; }
            }
            if (lane == 0) { wV[wid] = bv; wL[wid] = bl; wH[wid] = bh; wK[wid] = bk; }
            __syncthreads();
            if (tid == 0) {
                float v = wV[0]; unsigned l = wL[0], h = wH[0]; int k = wK[0];
                for (int w = 1; w < 8; ++w) {
                    if (cand_better(wV[w], wH[w], wL[w], wK[w], v, h, l, k)) {
                        v = wV[w]; l = wL[w]; h = wH[w]; k = wK[w];
                    }
                }
                selK[r] = k; selV[r] = v;
                sFs[k] = -INFINITY;   // remove from further rounds
            }
            __syncthreads();
        }

        // ---- state update + backpointer ----
        if (tid < BEAM_) {
            int k = selK[tid];
            int p = k / C_, c = k - p * C_;
            bool app = (c != BLANK_) && (sLen[p] == 0 || sLast[p] != c);
            unsigned long long h = sHH[k];
            nScore[tid] = selV[tid];
            nH1[tid] = (unsigned)h;
            nH2[tid] = (unsigned)(h >> 32);
            nLen[tid]  = sLen[p] + (app ? 1 : 0);
            nLast[tid] = app ? c : sLast[p];
            sBp[t * BEAM_ + tid] = (unsigned short)(p | ((app ? 1 : 0) << 4) | (c << 5));
        }
        __syncthreads();
        if (tid < BEAM_) {
            sScore[tid] = nScore[tid]; sH1[tid] = nH1[tid]; sH2[tid] = nH2[tid];
            sLen[tid] = nLen[tid]; sLast[tid] = nLast[tid];
        }
        __syncthreads();
    }

    // ---- outputs: tokens (B*T), then blen (B), then bscore (B), all as float ----
    if (tid == 0) {
        int best = 0; float bv = sScore[0];
        for (int i = 1; i < BEAM_; ++i) if (sScore[i] > bv) { bv = sScore[i]; best = i; }
        sBest = best; sBlen = sLen[best];
        out[(size_t)B_ * T_ + b]      = (float)sLen[best];
        out[(size_t)B_ * T_ + B_ + b] = sScore[best];
    }
    __syncthreads();

    float* trow = out + (size_t)b * T_;
    const int blen = sBlen;
    for (int i = tid; i < T_; i += NTHREADS)
        if (i >= blen) trow[i] = -1.0f;

    if (tid == 0) {   // backtrack through LDS backpointers
        int cur = sBest;
        int pos = blen;
        for (int t = len - 1; t >= 0; --t) {
            unsigned short bp = sBp[t * BEAM_ + cur];
            if ((bp >> 4) & 1) { --pos; trow[pos] = (float)(bp >> 5); }
            cur = bp & 15;
        }
    }
}

extern "C" void kernel_launch(void* const* d_in, const int* in_sizes, int n_in,
                              void* d_out, int out_size, void* d_ws, size_t ws_size,
                              hipStream_t stream) {
    (void)in_sizes; (void)n_in; (void)out_size; (void)d_ws; (void)ws_size;
    const float* logits  = (const float*)d_in[0];
    const int*   lengths = (const int*)d_in[1];
    float*       out     = (float*)d_out;
    ctc_beam_kernel<<<B_, NTHREADS, 0, stream>>>(logits, lengths, out);
}